// SNNLanguageModelGen2_55576876810901
// MI455X (gfx1250) — compile-verified
//
#include <hip/hip_runtime.h>

// ---------------------------------------------------------------------------
// SNN language model forward, MI455X (gfx1250), f32 WMMA 16x16x4 path with
// double-buffered GLOBAL_LOAD_ASYNC_TO_LDS staging (ASYNCcnt) when available.
// Shapes: T=512, B=8 -> TB=4096 rows; H=1024; M=32; VOCAB=32000.
// ---------------------------------------------------------------------------

#define TBROWS  4096
#define HDIM    1024
#define MDIM    32
#define TLEN    512
#define BSZ     8
#define NVOCAB  32000
#define KSUB    32

typedef __attribute__((ext_vector_type(2))) float v2f;
typedef __attribute__((ext_vector_type(4))) float f4;
typedef __attribute__((ext_vector_type(8))) float v8f;
typedef __attribute__((ext_vector_type(4))) int   v4i;

#if defined(__gfx1250__) && __has_builtin(__builtin_amdgcn_wmma_f32_16x16x4_f32)
#define HAVE_WMMA_F32 1
#else
#define HAVE_WMMA_F32 0
#endif

#if defined(__gfx1250__) && __has_builtin(__builtin_amdgcn_global_load_async_to_lds_b128)
#define HAVE_ASYNC_LDS 1
#else
#define HAVE_ASYNC_LDS 0
#endif

#if HAVE_ASYNC_LDS
typedef __attribute__((address_space(1))) v4i gv4i;   // global 128b payload
typedef __attribute__((address_space(3))) v4i lv4i;   // LDS 128b payload
#define AS1P(p) ((gv4i*)(p))
#define AS3P(p) ((lv4i*)(p))
__device__ __forceinline__ void wait_asynccnt0() {
#if __has_builtin(__builtin_amdgcn_s_wait_asynccnt)
  __builtin_amdgcn_s_wait_asynccnt(0);
#else
  asm volatile("s_wait_asynccnt 0x0" ::: "memory");
#endif
}
#endif

// ---------------------------------------------------------------------------
// Embedding gather: out[t,b,h] = emb[x[t,b],h] + pos[t,h]
// ---------------------------------------------------------------------------
__global__ __launch_bounds__(256) void embed_kernel(
    const int* __restrict__ x, const float* __restrict__ emb,
    const float* __restrict__ pos, float* __restrict__ out)
{
  int idx = blockIdx.x * 256 + threadIdx.x;     // < TBROWS*HDIM
  int h  = idx & (HDIM - 1);
  int tb = idx >> 10;
  int b  = tb & (BSZ - 1);
  int t  = tb >> 3;
  int tok = x[t * BSZ + b];
  out[idx] = emb[(size_t)tok * HDIM + h] + pos[t * HDIM + h];
}

// ---------------------------------------------------------------------------
// f32 WMMA GEMM: C[M,N] = A[M,K] * W[K,N] (+bias, +relu).
// Block = 256 threads (8 waves, wave32). Block tile 64 x BN, Ksub = 32.
// Wave (wm = w&3, wn = w>>2) owns rows [16*wm,16*wm+16), cols [wn*BN/2, +BN/2)
// as TN = BN/32 accumulators of 16x16.
// Async path: double-buffered LDS; slab k0+KSUB streams via
// global_load_async_to_lds_b128 while slab k0 is consumed by WMMA.
// ---------------------------------------------------------------------------
template<int BN, bool BIAS, bool RELU>
__global__ __launch_bounds__(256) void gemm_f32_wmma(
    const float* __restrict__ A, const float* __restrict__ W,
    const float* __restrict__ bias, float* __restrict__ C,
    int M, int N, int K)
{
  constexpr int TN = BN / 32;
  // Row strides: As 36 floats (144B, 16B-aligned granules, conflict-free column
  // reads: 36*m mod 64 distinct), Bs BN+4 floats (272B for BN=64).
  __shared__ float As[2][64][KSUB + 4];
  __shared__ float Bs[2][KSUB][BN + 4];

  const int tid  = threadIdx.x;
  const int lane = tid & 31;
  const int wave = tid >> 5;
  const int wm   = wave & 3;
  const int wn   = wave >> 2;
  const int l16  = lane & 15;
  const int lh   = lane >> 4;
  const int mbase = wm * 16;
  const int nbase = wn * (BN / 2);
  const int bm = blockIdx.x;
  const int bn = blockIdx.y;

  v8f acc[TN];
  for (int t = 0; t < TN; ++t)
    for (int r = 0; r < 8; ++r) acc[t][r] = 0.0f;

  const int AG = (64 * KSUB) / 4;      // 512 f4 granules
  const int BG = (KSUB * BN) / 4;      // 512 (BN=64) or 256 (BN=32)

#if HAVE_ASYNC_LDS
  // ---- async double-buffered pipeline -------------------------------------
  auto issue_tile = [&](int k0, int buf) {
    for (int g = tid; g < AG; g += 256) {
      int ar = g >> 3, ac = (g & 7) * 4;
      __builtin_amdgcn_global_load_async_to_lds_b128(
          AS1P(&A[(size_t)(bm * 64 + ar) * K + k0 + ac]),
          AS3P(&As[buf][ar][ac]), 0, 0);
    }
    for (int g = tid; g < BG; g += 256) {
      int br = g / (BN / 4), bc = (g % (BN / 4)) * 4;
      __builtin_amdgcn_global_load_async_to_lds_b128(
          AS1P(&W[(size_t)(k0 + br) * N + bn * BN + bc]),
          AS3P(&Bs[buf][br][bc]), 0, 0);
    }
  };

  issue_tile(0, 0);
  wait_asynccnt0();
  __syncthreads();

  for (int k0 = 0; k0 < K; k0 += KSUB) {
    const int cur = (k0 / KSUB) & 1;
    if (k0 + KSUB < K) issue_tile(k0 + KSUB, cur ^ 1);  // fill other buffer

    for (int kb = 0; kb < KSUB; kb += 4) {
#if HAVE_WMMA_F32
      const int kfr = kb + lh * 2;
      v2f afr; afr[0] = As[cur][mbase + l16][kfr]; afr[1] = As[cur][mbase + l16][kfr + 1];
      for (int t = 0; t < TN; ++t) {
        v2f bfr;
        bfr[0] = Bs[cur][kfr][nbase + t * 16 + l16];
        bfr[1] = Bs[cur][kfr + 1][nbase + t * 16 + l16];
        acc[t] = __builtin_amdgcn_wmma_f32_16x16x4_f32(
            false, afr, false, bfr, (short)0, acc[t], false, false);
      }
#else
      for (int t = 0; t < TN; ++t)
        for (int r = 0; r < 8; ++r) {
          float s = acc[t][r];
          for (int kk = 0; kk < 4; ++kk)
            s = fmaf(As[cur][mbase + r + lh * 8][kb + kk],
                     Bs[cur][kb + kk][nbase + t * 16 + l16], s);
          acc[t][r] = s;
        }
#endif
    }

    wait_asynccnt0();   // next buffer fully in LDS (this wave's issues)
    __syncthreads();    // ...and every other wave's too
  }
#else
  // ---- synchronous fallback (single buffer + prefetch hints) --------------
  for (int k0 = 0; k0 < K; k0 += KSUB) {
    __syncthreads();
    for (int g = tid; g < AG; g += 256) {
      int ar = g >> 3, ac = (g & 7) * 4;
      f4 v = *(const f4*)(&A[(size_t)(bm * 64 + ar) * K + k0 + ac]);
      *(f4*)(&As[0][ar][ac]) = v;
    }
    for (int g = tid; g < BG; g += 256) {
      int br = g / (BN / 4), bc = (g % (BN / 4)) * 4;
      f4 v = *(const f4*)(&W[(size_t)(k0 + br) * N + bn * BN + bc]);
      *(f4*)(&Bs[0][br][bc]) = v;
    }
    if (k0 + KSUB < K) {
      __builtin_prefetch(&A[(size_t)(bm * 64 + (tid >> 3)) * K + k0 + KSUB + (tid & 7) * 4], 0, 1);
      __builtin_prefetch(&W[(size_t)(k0 + KSUB + tid / (BN / 4)) * N + bn * BN + (tid % (BN / 4)) * 4], 0, 1);
    }
    __syncthreads();

    for (int kb = 0; kb < KSUB; kb += 4) {
#if HAVE_WMMA_F32
      const int kfr = kb + lh * 2;
      v2f afr; afr[0] = As[0][mbase + l16][kfr]; afr[1] = As[0][mbase + l16][kfr + 1];
      for (int t = 0; t < TN; ++t) {
        v2f bfr;
        bfr[0] = Bs[0][kfr][nbase + t * 16 + l16];
        bfr[1] = Bs[0][kfr + 1][nbase + t * 16 + l16];
        acc[t] = __builtin_amdgcn_wmma_f32_16x16x4_f32(
            false, afr, false, bfr, (short)0, acc[t], false, false);
      }
#else
      for (int t = 0; t < TN; ++t)
        for (int r = 0; r < 8; ++r) {
          float s = acc[t][r];
          for (int kk = 0; kk < 4; ++kk)
            s = fmaf(As[0][mbase + r + lh * 8][kb + kk],
                     Bs[0][kb + kk][nbase + t * 16 + l16], s);
          acc[t][r] = s;
        }
#endif
    }
  }
#endif

  // Epilogue. C/D layout: reg r, lanes 0-15 -> M=r, lanes 16-31 -> M=r+8.
  for (int t = 0; t < TN; ++t) {
    const int ncol = bn * BN + nbase + t * 16 + l16;
    float bv = BIAS ? bias[ncol] : 0.0f;
    for (int r = 0; r < 8; ++r) {
      int mrow = bm * 64 + mbase + r + lh * 8;
      float v = acc[t][r] + bv;
      if (RELU) v = fmaxf(v, 0.0f);
      C[(size_t)mrow * N + ncol] = v;
    }
  }
}

// ---------------------------------------------------------------------------
// tau-softmax over M=32: one wave32 per row. Rows [0,4096) -> q/2 -> a,
// rows [4096,8192) -> k/2 -> b. Grid 1024 x 256 (8 waves/block).
// ---------------------------------------------------------------------------
__global__ __launch_bounds__(256) void softmax32_kernel(
    const float* __restrict__ q, const float* __restrict__ k,
    float* __restrict__ a, float* __restrict__ bsm)
{
  const int lane = threadIdx.x & 31;
  const int wave = threadIdx.x >> 5;
  int row = blockIdx.x * 8 + wave;              // 0..8191
  const float* src; float* dst;
  if (row < TBROWS) { src = q; dst = a; }
  else              { src = k; dst = bsm; row -= TBROWS; }
  float v = src[row * MDIM + lane] * 0.5f;      // tau = 2.0
  float mx = v;
  for (int off = 16; off >= 1; off >>= 1) mx = fmaxf(mx, __shfl_xor(mx, off, 32));
  float e = expf(v - mx);
  float s = e;
  for (int off = 16; off >= 1; off >>= 1) s += __shfl_xor(s, off, 32);
  dst[row * MDIM + lane] = e / s;
}

// ---------------------------------------------------------------------------
// Leaky associative scan. One thread per (b,i,j) mem cell, sequential over T.
// Forward value of spk_st is exactly the binary spike. 8192 threads total.
// ---------------------------------------------------------------------------
__global__ __launch_bounds__(256) void scan_kernel(
    const float* __restrict__ a, const float* __restrict__ bsm,
    float* __restrict__ spk)
{
  int cell = blockIdx.x * 256 + threadIdx.x;    // < 8192
  int b  = cell >> 10;
  int ij = cell & 1023;
  int i = ij >> 5, j = ij & 31;
  float mem = 0.0f;
  for (int t = 0; t < TLEN; ++t) {
    int rb = t * BSZ + b;
    mem = fmaf(0.9f, mem, a[rb * MDIM + i] * bsm[rb * MDIM + j]);  // BETA=0.9
    float s = (mem > 1.0f) ? 1.0f : 0.0f;                          // THR=1.0
    mem -= s;                                                      // soft reset
    spk[(size_t)rb * HDIM + ij] = s;
  }
}

// ---------------------------------------------------------------------------
// LayerNorm over H=1024 (input already bias+ReLU'd by GEMM epilogue).
// One 256-thread block per row; wave32 shuffle + LDS cross-wave reduce.
// ---------------------------------------------------------------------------
__global__ __launch_bounds__(256) void layernorm_kernel(
    float* __restrict__ h, const float* __restrict__ g, const float* __restrict__ be)
{
  const int row = blockIdx.x;
  const int tid = threadIdx.x;
  float v[4]; float s = 0.0f, sq = 0.0f;
  for (int i = 0; i < 4; ++i) {
    v[i] = h[(size_t)row * HDIM + tid + i * 256];
    s += v[i]; sq += v[i] * v[i];
  }
  for (int off = 16; off >= 1; off >>= 1) {
    s  += __shfl_xor(s,  off, 32);
    sq += __shfl_xor(sq, off, 32);
  }
  __shared__ float ss[8], ssq[8];
  if ((tid & 31) == 0) { ss[tid >> 5] = s; ssq[tid >> 5] = sq; }
  __syncthreads();
  s = 0.0f; sq = 0.0f;
  for (int w = 0; w < 8; ++w) { s += ss[w]; sq += ssq[w]; }
  const float mean = s * (1.0f / HDIM);
  const float var  = sq * (1.0f / HDIM) - mean * mean;
  const float rinv = rsqrtf(var + 1e-5f);
  for (int i = 0; i < 4; ++i) {
    int c = tid + i * 256;
    h[(size_t)row * HDIM + c] = (v[i] - mean) * rinv * g[c] + be[c];
  }
}

// ---------------------------------------------------------------------------
// Launch: embed -> [assoc -> GEMM(+bias,relu) -> LN] x2 -> assoc -> GEMM(relu)
//         -> logits GEMM (streams 524MB to d_out; the HBM-bound stage).
// ---------------------------------------------------------------------------
extern "C" void kernel_launch(void* const* d_in, const int* in_sizes, int n_in,
                              void* d_out, int out_size, void* d_ws, size_t ws_size,
                              hipStream_t stream)
{
  (void)in_sizes; (void)n_in; (void)out_size; (void)ws_size;
  const int*   x    = (const int*)  d_in[0];
  const float* emb  = (const float*)d_in[1];
  const float* pos  = (const float*)d_in[2];
  const float* Wq[3]= {(const float*)d_in[3], (const float*)d_in[5], (const float*)d_in[7]};
  const float* Wk[3]= {(const float*)d_in[4], (const float*)d_in[6], (const float*)d_in[8]};
  const float* W2   = (const float*)d_in[9];  const float* b2 = (const float*)d_in[10];
  const float* W3   = (const float*)d_in[11]; const float* b3 = (const float*)d_in[12];
  const float* W4   = (const float*)d_in[13]; const float* b4 = (const float*)d_in[14];
  const float* Wout = (const float*)d_in[15]; const float* bout = (const float*)d_in[16];
  const float* lng[2] = {(const float*)d_in[17], (const float*)d_in[19]};
  const float* lnb[2] = {(const float*)d_in[18], (const float*)d_in[20]};
  float* out = (float*)d_out;

  // Workspace layout (~35.7 MB)
  float* hA = (float*)d_ws;
  float* hB = hA + (size_t)TBROWS * HDIM;
  float* qb = hB + (size_t)TBROWS * HDIM;
  float* kb = qb + (size_t)TBROWS * MDIM;
  float* ab = kb + (size_t)TBROWS * MDIM;
  float* bb = ab + (size_t)TBROWS * MDIM;

  auto assoc = [&](const float* xin, const float* wq, const float* wk, float* spk_out) {
    dim3 gqk(TBROWS / 64, MDIM / 32);  // (64, 1)
    gemm_f32_wmma<32, false, false><<<gqk, 256, 0, stream>>>(xin, wq, nullptr, qb, TBROWS, MDIM, HDIM);
    gemm_f32_wmma<32, false, false><<<gqk, 256, 0, stream>>>(xin, wk, nullptr, kb, TBROWS, MDIM, HDIM);
    softmax32_kernel<<<1024, 256, 0, stream>>>(qb, kb, ab, bb);
    scan_kernel<<<32, 256, 0, stream>>>(ab, bb, spk_out);
  };

  embed_kernel<<<(TBROWS * HDIM) / 256, 256, 0, stream>>>(x, emb, pos, hA);

  assoc(hA, Wq[0], Wk[0], hB);
  gemm_f32_wmma<64, true, true><<<dim3(TBROWS / 64, HDIM / 64), 256, 0, stream>>>(hB, W2, b2, hA, TBROWS, HDIM, HDIM);
  layernorm_kernel<<<TBROWS, 256, 0, stream>>>(hA, lng[0], lnb[0]);

  assoc(hA, Wq[1], Wk[1], hB);
  gemm_f32_wmma<64, true, true><<<dim3(TBROWS / 64, HDIM / 64), 256, 0, stream>>>(hB, W3, b3, hA, TBROWS, HDIM, HDIM);
  layernorm_kernel<<<TBROWS, 256, 0, stream>>>(hA, lng[1], lnb[1]);

  assoc(hA, Wq[2], Wk[2], hB);
  gemm_f32_wmma<64, true, true><<<dim3(TBROWS / 64, HDIM / 64), 256, 0, stream>>>(hB, W4, b4, hA, TBROWS, HDIM, HDIM);

  gemm_f32_wmma<64, true, false><<<dim3(TBROWS / 64, NVOCAB / 64), 256, 0, stream>>>(hA, Wout, bout, out, TBROWS, NVOCAB, HDIM);
}